// FullQuantumModel_52510270161290
// MI455X (gfx1250) — compile-verified
//
#include <hip/hip_runtime.h>

// Pipeline: state lives in d_ws between kernels (requires ws_size >= 128MB + 4KB;
// state is 256 x 65536 complex64 as separate float re/im planes -> L2-resident).
#define TPB   256
#define AMPS  64
#define NQ    16

typedef float v2f __attribute__((ext_vector_type(2)));
typedef float v8f __attribute__((ext_vector_type(8)));

// ---------------- rotation on a register-local amplitude bit Q (bits 0..5) ----------------
template<int Q>
__device__ __forceinline__ void rot_local(float (&re)[AMPS], float (&im)[AMPS],
                                          const float* u) {
  const float u00r=u[0], u00i=u[1], u01r=u[2], u01i=u[3];
  const float u10r=u[4], u10i=u[5], u11r=u[6], u11i=u[7];
#pragma unroll
  for (int k = 0; k < AMPS; ++k) {
    if (k & (1 << Q)) continue;
    const int k2 = k | (1 << Q);
    const float a0r=re[k], a0i=im[k], a1r=re[k2], a1i=im[k2];
    re[k]  = u00r*a0r - u00i*a0i + u01r*a1r - u01i*a1i;
    im[k]  = u00r*a0i + u00i*a0r + u01r*a1i + u01i*a1r;
    re[k2] = u10r*a0r - u10i*a0i + u11r*a1r - u11i*a1i;
    im[k2] = u10r*a0i + u10i*a0r + u11r*a1i + u11i*a1r;
  }
}

// ---------------- rotation on a lane bit (amplitude bits 6..10 -> lane bits 0..4) ----------
template<int M>
__device__ __forceinline__ void rot_lane(float (&re)[AMPS], float (&im)[AMPS],
                                         const float* u, int lane) {
  const float u00r=u[0], u00i=u[1], u01r=u[2], u01i=u[3];
  const float u10r=u[4], u10i=u[5], u11r=u[6], u11i=u[7];
  const bool hi = (lane & M) != 0;
  const float xr = hi ? u11r : u00r, xi = hi ? u11i : u00i;
  const float yr = hi ? u10r : u01r, yi = hi ? u10i : u01i;
#pragma unroll
  for (int k = 0; k < AMPS; ++k) {
    const float pr = __shfl_xor(re[k], M, 32);
    const float pi = __shfl_xor(im[k], M, 32);
    const float ar = re[k], ai = im[k];
    re[k] = xr*ar - xi*ai + yr*pr - yi*pi;
    im[k] = xr*ai + xi*ar + yr*pi + yi*pr;
  }
}

// ------ rotation on a wave bit (thread bits 5..7), single full-tile LDS round --------------
template<int TX>
__device__ __forceinline__ void rot_wave(float (&re)[AMPS], float (&im)[AMPS],
                                         const float* u, int tid, float2* sb) {
  const float u00r=u[0], u00i=u[1], u01r=u[2], u01i=u[3];
  const float u10r=u[4], u10i=u[5], u11r=u[6], u11i=u[7];
  const bool hi = (tid & TX) != 0;
  const float xr = hi ? u11r : u00r, xi = hi ? u11i : u00i;
  const float yr = hi ? u10r : u01r, yi = hi ? u10i : u01i;
  const int partner = tid ^ TX;
  __syncthreads();
#pragma unroll
  for (int k = 0; k < AMPS; ++k) sb[k*TPB + tid] = make_float2(re[k], im[k]);
  __syncthreads();
#pragma unroll
  for (int k = 0; k < AMPS; ++k) {
    const float2 p = sb[k*TPB + partner];
    const float ar = re[k], ai = im[k];
    re[k] = xr*ar - xi*ai + yr*p.x - yi*p.y;
    im[k] = xr*ai + xi*ar + yr*p.y + yi*p.x;
  }
}

// ---------------- CNOT: control and target both register-local -----------------------------
template<int C,int T>
__device__ __forceinline__ void cnot_ll(float (&re)[AMPS], float (&im)[AMPS]) {
#pragma unroll
  for (int k = 0; k < AMPS; ++k) {
    if ((k & (1 << C)) && !(k & (1 << T))) {
      const int k2 = k | (1 << T);
      float t;
      t = re[k]; re[k] = re[k2]; re[k2] = t;
      t = im[k]; im[k] = im[k2]; im[k2] = t;
    }
  }
}

// ---------------- CNOT: control bit in tid, target a lane bit (shuffle swap) ---------------
template<int CM,int TM>
__device__ __forceinline__ void cnot_lane(float (&re)[AMPS], float (&im)[AMPS], int tid) {
  const bool cb = (tid & CM) != 0;
#pragma unroll
  for (int k = 0; k < AMPS; ++k) {
    const float pr = __shfl_xor(re[k], TM, 32);
    const float pi = __shfl_xor(im[k], TM, 32);
    re[k] = cb ? pr : re[k];
    im[k] = cb ? pi : im[k];
  }
}

// ---------------- CNOT: control bit in tid, target register-local --------------------------
template<int CM,int T>
__device__ __forceinline__ void cnot_glocal(float (&re)[AMPS], float (&im)[AMPS], int tid) {
  const bool cb = (tid & CM) != 0;
#pragma unroll
  for (int k = 0; k < AMPS; ++k) {
    if (!(k & (1 << T))) {
      const int k2 = k | (1 << T);
      const float r0 = re[k], r1 = re[k2], i0 = im[k], i1 = im[k2];
      re[k]  = cb ? r1 : r0;  re[k2] = cb ? r0 : r1;
      im[k]  = cb ? i1 : i0;  im[k2] = cb ? i0 : i1;
    }
  }
}

// ---------------- CNOT: control bit in tid, target a wave bit (LDS exchange) ---------------
template<int CM,int TX>
__device__ __forceinline__ void cnot_wave(float (&re)[AMPS], float (&im)[AMPS], int tid,
                                          float2* sb) {
  const bool cb = (tid & CM) != 0;
  const int src = cb ? (tid ^ TX) : tid;
  __syncthreads();
#pragma unroll
  for (int k = 0; k < AMPS; ++k) sb[k*TPB + tid] = make_float2(re[k], im[k]);
  __syncthreads();
#pragma unroll
  for (int k = 0; k < AMPS; ++k) {
    const float2 p = sb[k*TPB + src];
    re[k] = p.x; im[k] = p.y;
  }
}

// ----- wrap CNOT(15,0): control amp bit0 (odd k), target amp bit15 (tid bit7 in CB tiling) -
__device__ __forceinline__ void wrap_cnot(float (&re)[AMPS], float (&im)[AMPS], int tid,
                                          float2* sb) {
  __syncthreads();
#pragma unroll
  for (int k = 1; k < AMPS; k += 2) sb[k*TPB + tid] = make_float2(re[k], im[k]);
  __syncthreads();
#pragma unroll
  for (int k = 1; k < AMPS; k += 2) {
    const float2 p = sb[k*TPB + (tid ^ 128)];
    re[k] = p.x; im[k] = p.y;
  }
}

// ---------------- per-layer gate matrices: U = RZ(g) RY(b) RX(a), half angles --------------
__device__ __forceinline__ void build_gates(float (*ug)[8], const float* par, int layer, int tid) {
  if (tid < NQ) {
    const float* p = par + (layer*NQ + tid)*3;
    float sa, ca, sb, cb, sg, cg;
    sincosf(0.5f*p[0], &sa, &ca);
    sincosf(0.5f*p[1], &sb, &cb);
    sincosf(0.5f*p[2], &sg, &cg);
    const float m00r= cb*ca, m00i= sb*sa;
    const float m01r=-sb*ca, m01i=-cb*sa;
    const float m10r= sb*ca, m10i=-cb*sa;
    const float m11r= cb*ca, m11i=-sb*sa;
    ug[tid][0] = cg*m00r + sg*m00i;  ug[tid][1] = cg*m00i - sg*m00r;
    ug[tid][2] = cg*m01r + sg*m01i;  ug[tid][3] = cg*m01i - sg*m01r;
    ug[tid][4] = cg*m10r - sg*m10i;  ug[tid][5] = cg*m10i + sg*m10r;
    ug[tid][6] = cg*m11r - sg*m11i;  ug[tid][7] = cg*m11i + sg*m11r;
  }
}

// ============ CB kernel: tile bits {0..10,13,14,15}; WG id carries bits 11,12 ==============
// ops: [wrap CNOT prev layer], R(bits0..3) via WMMA, R(bits4..10,13,14,15),
//      CNOT(0,1)=ctrl b15->tgt b14, CNOT(1,2)=ctrl b14->tgt b13
extern "C" __global__ __launch_bounds__(TPB, 1)
void qsim_cb(const float* __restrict__ src_re, const float* __restrict__ src_im,
             float* __restrict__ dst_re, float* __restrict__ dst_im,
             const float* __restrict__ par, int layer, int do_wrap)
{
  __shared__ float  ug[NQ][8];
  __shared__ float  wre[16][16], wim[16][16];
  __shared__ float2 sbuf[AMPS*TPB];    // 128 KB

  const int tid = threadIdx.x, lane = tid & 31;
  const int wg = blockIdx.x;
  const size_t base = ((size_t)(wg >> 2) << 16)
                    | ((size_t)((tid >> 7) & 1) << 15)
                    | ((size_t)((tid >> 6) & 1) << 14)
                    | ((size_t)((tid >> 5) & 1) << 13)
                    | ((size_t)((wg >> 1) & 1) << 12)
                    | ((size_t)(wg & 1) << 11)
                    | ((size_t)(tid & 31) << 6);

  __builtin_prefetch(src_re + base, 0, 3);   // global_prefetch_b8
  __builtin_prefetch(src_im + base, 0, 3);

  build_gates(ug, par, layer, tid);
  __syncthreads();

  // W = U(b3) (x) U(b2) (x) U(b1) (x) U(b0); amp bit i -> qubit 15-i -> ug[15-i]
  {
    const int r = tid >> 4, c = tid & 15;
    float cr = 1.0f, ci = 0.0f;
#pragma unroll
    for (int i = 0; i < 4; ++i) {
      const int ri = (r >> i) & 1, cc = (c >> i) & 1;
      const float fr = ug[15-i][(ri*2+cc)*2], fi = ug[15-i][(ri*2+cc)*2+1];
      const float nr = cr*fr - ci*fi, ni = cr*fi + ci*fr;
      cr = nr; ci = ni;
    }
    wre[r][c] = cr; wim[r][c] = ci;
  }

  float re[AMPS], im[AMPS];
  {
    const float4* pr = (const float4*)(src_re + base);
    const float4* pi = (const float4*)(src_im + base);
#pragma unroll
    for (int v = 0; v < AMPS/4; ++v) {
      const float4 r = pr[v], i4 = pi[v];
      re[4*v+0]=r.x;  re[4*v+1]=r.y;  re[4*v+2]=r.z;  re[4*v+3]=r.w;
      im[4*v+0]=i4.x; im[4*v+1]=i4.y; im[4*v+2]=i4.z; im[4*v+3]=i4.w;
    }
  }
  __syncthreads();   // wre/wim visible

  if (do_wrap) wrap_cnot(re, im, tid, sbuf);   // CNOT(15,0) of previous layer

  // ---- WMMA stage: fused rotations on amp bits 0..3, D = W * S (complex) ----
  // All stage LDS traffic is wave-private; D overwrites S in place per tile.
  __syncthreads();
#pragma unroll
  for (int k = 0; k < AMPS; ++k)
    sbuf[k*TPB + (tid ^ ((k >> 4) << 3))] = make_float2(re[k], im[k]);
  __syncthreads();
  {
    const int m = lane & 15, h = lane >> 4;
    const int wbase = tid & ~31;
    v2f aRe[4], aIm[4];
#pragma unroll
    for (int kc = 0; kc < 4; ++kc) {
      aRe[kc][0] = wre[m][4*kc + 2*h];  aRe[kc][1] = wre[m][4*kc + 2*h + 1];
      aIm[kc][0] = wim[m][4*kc + 2*h];  aIm[kc][1] = wim[m][4*kc + 2*h + 1];
    }
#pragma unroll 1
    for (int ct = 0; ct < 8; ++ct) {
      v2f bRe[4], bIm[4];
#pragma unroll
      for (int kc = 0; kc < 4; ++kc) {
#pragma unroll
        for (int j = 0; j < 2; ++j) {
          const int row = 4*kc + 2*h + j;          // K index of B row
          const int c   = ct*16 + m;               // column (amp bits 4..10)
          const int tp  = wbase + (c >> 2);        // holder thread
          const int kp  = ((c & 3) << 4) | row;    // holder slot
          const float2 v = sbuf[kp*TPB + (tp ^ ((kp >> 4) << 3))];
          bRe[kc][j] = v.x; bIm[kc][j] = v.y;
        }
      }
      v8f aRR = {0.f,0.f,0.f,0.f,0.f,0.f,0.f,0.f};
      v8f aII = aRR, aRI = aRR, aIR = aRR;
#pragma unroll
      for (int kc = 0; kc < 4; ++kc) {
        aRR = __builtin_amdgcn_wmma_f32_16x16x4_f32(false, aRe[kc], false, bRe[kc], (short)0, aRR, false, false);
        aII = __builtin_amdgcn_wmma_f32_16x16x4_f32(false, aIm[kc], false, bIm[kc], (short)0, aII, false, false);
        aRI = __builtin_amdgcn_wmma_f32_16x16x4_f32(false, aRe[kc], false, bIm[kc], (short)0, aRI, false, false);
        aIR = __builtin_amdgcn_wmma_f32_16x16x4_f32(false, aIm[kc], false, bRe[kc], (short)0, aIR, false, false);
      }
#pragma unroll
      for (int r = 0; r < 8; ++r) {
        const int mm = r + 8*h;
        const int c  = ct*16 + m;
        const int tp = wbase + (c >> 2);
        const int kp = ((c & 3) << 4) | mm;
        sbuf[kp*TPB + (tp ^ ((kp >> 4) << 3))] = make_float2(aRR[r]-aII[r], aRI[r]+aIR[r]);
      }
    }
  }
  __syncthreads();
#pragma unroll
  for (int k = 0; k < AMPS; ++k) {
    const float2 v = sbuf[k*TPB + (tid ^ ((k >> 4) << 3))];
    re[k] = v.x; im[k] = v.y;
  }

  // remaining rotations of this layer
  rot_local<4>(re, im, ug[11]);
  rot_local<5>(re, im, ug[10]);
  rot_lane<1>(re, im, ug[9], lane);
  rot_lane<2>(re, im, ug[8], lane);
  rot_lane<4>(re, im, ug[7], lane);
  rot_lane<8>(re, im, ug[6], lane);
  rot_lane<16>(re, im, ug[5], lane);
  rot_wave<32>(re, im, ug[2], tid, sbuf);    // amp bit13 (qubit 2)
  rot_wave<64>(re, im, ug[1], tid, sbuf);    // amp bit14 (qubit 1)
  rot_wave<128>(re, im, ug[0], tid, sbuf);   // amp bit15 (qubit 0)
  cnot_wave<128,64>(re, im, tid, sbuf);      // CNOT(0,1): ctrl b15, tgt b14
  cnot_wave<64,32>(re, im, tid, sbuf);       // CNOT(1,2): ctrl b14, tgt b13

  {
    float4* pr = (float4*)(dst_re + base);
    float4* pi = (float4*)(dst_im + base);
#pragma unroll
    for (int v = 0; v < AMPS/4; ++v) {
      pr[v] = make_float4(re[4*v+0], re[4*v+1], re[4*v+2], re[4*v+3]);
      pi[v] = make_float4(im[4*v+0], im[4*v+1], im[4*v+2], im[4*v+3]);
    }
  }
}

// ============ A kernel: tile bits {0..13} contiguous; WG id carries bits 14,15 =============
// ops: R(bit11), R(bit12), CNOT(2,3)..CNOT(14,15)
extern "C" __global__ __launch_bounds__(TPB, 1)
void qsim_a(float* __restrict__ st_re, float* __restrict__ st_im,
            const float* __restrict__ par, int layer)
{
  __shared__ float  ug[NQ][8];
  __shared__ float2 sbuf[AMPS*TPB];

  const int tid = threadIdx.x, lane = tid & 31;
  const size_t base = ((size_t)blockIdx.x << 14) | ((size_t)tid << 6);

  __builtin_prefetch(st_re + base, 0, 3);
  __builtin_prefetch(st_im + base, 0, 3);

  build_gates(ug, par, layer, tid);

  float re[AMPS], im[AMPS];
  {
    const float4* pr = (const float4*)(st_re + base);
    const float4* pi = (const float4*)(st_im + base);
#pragma unroll
    for (int v = 0; v < AMPS/4; ++v) {
      const float4 r = pr[v], i4 = pi[v];
      re[4*v+0]=r.x;  re[4*v+1]=r.y;  re[4*v+2]=r.z;  re[4*v+3]=r.w;
      im[4*v+0]=i4.x; im[4*v+1]=i4.y; im[4*v+2]=i4.z; im[4*v+3]=i4.w;
    }
  }
  __syncthreads();

  rot_wave<32>(re, im, ug[4], tid, sbuf);    // amp bit11 (qubit 4)
  rot_wave<64>(re, im, ug[3], tid, sbuf);    // amp bit12 (qubit 3)
  cnot_wave<128,64>(re, im, tid, sbuf);      // CNOT(2,3):  ctrl b13, tgt b12
  cnot_wave<64,32>(re, im, tid, sbuf);       // CNOT(3,4):  ctrl b12, tgt b11
  cnot_lane<32,16>(re, im, tid);             // CNOT(4,5):  ctrl b11, tgt b10
  cnot_lane<16,8>(re, im, tid);              // CNOT(5,6)
  cnot_lane<8,4>(re, im, tid);               // CNOT(6,7)
  cnot_lane<4,2>(re, im, tid);               // CNOT(7,8)
  cnot_lane<2,1>(re, im, tid);               // CNOT(8,9)
  cnot_glocal<1,5>(re, im, tid);             // CNOT(9,10): ctrl b6, tgt b5
  cnot_ll<5,4>(re, im);                      // CNOT(10,11)
  cnot_ll<4,3>(re, im);                      // CNOT(11,12)
  cnot_ll<3,2>(re, im);                      // CNOT(12,13)
  cnot_ll<2,1>(re, im);                      // CNOT(13,14)
  cnot_ll<1,0>(re, im);                      // CNOT(14,15)

  {
    float4* pr = (float4*)(st_re + base);
    float4* pi = (float4*)(st_im + base);
#pragma unroll
    for (int v = 0; v < AMPS/4; ++v) {
      pr[v] = make_float4(re[4*v+0], re[4*v+1], re[4*v+2], re[4*v+3]);
      pi[v] = make_float4(im[4*v+0], im[4*v+1], im[4*v+2], im[4*v+3]);
    }
  }
}

// ===== F: final wrap CNOT folded into parity selection + per-WG |amp|^2 partial sums =======
extern "C" __global__ __launch_bounds__(TPB, 1)
void qsim_f(const float* __restrict__ st_re, const float* __restrict__ st_im,
            float* __restrict__ partial)
{
  __shared__ float rs[TPB/32];
  const int tid = threadIdx.x, lane = tid & 31, wv = tid >> 5;
  const size_t base = ((size_t)blockIdx.x << 14) | ((size_t)tid << 6);
  const int p = (blockIdx.x >> 1) & 1;    // amp bit15 of this tile
  float s = 0.0f;
#pragma unroll
  for (int k = 0; k < AMPS; ++k) {
    if ((k & 1) == p) {   // post-wrap bit15==0 amps: bit0 == bit15 (pre-wrap)
      const float r = st_re[base + k], i = st_im[base + k];
      s += r*r + i*i;
    }
  }
#pragma unroll
  for (int o = 16; o > 0; o >>= 1) s += __shfl_xor(s, o, 32);
  if (lane == 0) rs[wv] = s;
  __syncthreads();
  if (tid == 0) {
    float t = 0.0f;
#pragma unroll
    for (int w = 0; w < TPB/32; ++w) t += rs[w];
    partial[blockIdx.x] = t;
  }
}

extern "C" __global__ void qsim_g(const float* __restrict__ partial, float* __restrict__ out) {
  const int t = threadIdx.x;
  out[t] = partial[4*t] + partial[4*t+1] + partial[4*t+2] + partial[4*t+3];
}

extern "C" void kernel_launch(void* const* d_in, const int* in_sizes, int n_in,
                              void* d_out, int out_size, void* d_ws, size_t ws_size,
                              hipStream_t stream) {
  (void)in_sizes; (void)n_in; (void)ws_size;
  const float* sre = (const float*)d_in[0];
  const float* sim = (const float*)d_in[1];
  const float* par = (const float*)d_in[2];
  float* out = (float*)d_out;

  const size_t N = (size_t)out_size * 65536;       // amps total (256 * 2^16)
  float* wsre    = (float*)d_ws;
  float* wsim    = wsre + N;
  float* partial = wsim + N;
  const unsigned nblk = (unsigned)out_size * 4;    // 4 WGs per batch element

  // layer 0: read input, write workspace
  qsim_cb<<<dim3(nblk), dim3(TPB), 0, stream>>>(sre, sim, wsre, wsim, par, 0, 0);
  qsim_a <<<dim3(nblk), dim3(TPB), 0, stream>>>(wsre, wsim, par, 0);
  // layers 1..3: in-place, CB applies previous layer's wrap CNOT first
  for (int l = 1; l < 4; ++l) {
    qsim_cb<<<dim3(nblk), dim3(TPB), 0, stream>>>(wsre, wsim, wsre, wsim, par, l, 1);
    qsim_a <<<dim3(nblk), dim3(TPB), 0, stream>>>(wsre, wsim, par, l);
  }
  // final wrap folded into parity trick + reduction
  qsim_f<<<dim3(nblk), dim3(TPB), 0, stream>>>(wsre, wsim, partial);
  qsim_g<<<dim3(1), dim3((unsigned)out_size), 0, stream>>>(partial, out);
}